// SynapsePredictor_32607391711951
// MI455X (gfx1250) — compile-verified
//
#include <hip/hip_runtime.h>
#include <hip/hip_bf16.h>

// CDNA5 / gfx1250, wave32. Matrix ops use V_WMMA_F32_16X16X4_F32 (f32 in/out,
// precision-safe vs the f32 reference; GEMM flops are tiny vs the memory-bound
// gather/scatter so the f32 WMMA rate is irrelevant to the roofline).

typedef __attribute__((ext_vector_type(2))) float v2f_t;
typedef __attribute__((ext_vector_type(8))) float v8f_t;

#define DIM 128         // IN_CH == HID == 128
#define LDS_STRIDE 132  // 128 + 4 pad: ds_load_b64 reads hit bank pair {4m,4m+1}

// ---------------------------------------------------------------------------
// Edge scatter: agg[dst] += x[src] * w  (atomic f32, L2-resident accumulator).
// One edge per wave (32 lanes x 4 floats = 128 features), coalesced loads and
// coalesced atomic bursts. Optionally counts in-degree once per edge.
// ---------------------------------------------------------------------------
__global__ void scatter_edges(const float* __restrict__ feat,
                              const int* __restrict__ edge_index,
                              const float* __restrict__ ewt,
                              float* __restrict__ agg,
                              float* __restrict__ deg,
                              int E, int count_deg)
{
    long long tid = (long long)blockIdx.x * blockDim.x + threadIdx.x;
    if (tid >= (long long)E * 32) return;
    int e = (int)(tid >> 5);
    int c = (int)(tid & 31);          // 4-float feature chunk
    int s = edge_index[e];
    int d = edge_index[E + e];
    float w = ewt[e];
    const float4 xv = *(const float4*)(feat + (size_t)s * DIM + c * 4);
    float* dp = agg + (size_t)d * DIM + c * 4;
    atomicAdd(dp + 0, xv.x * w);
    atomicAdd(dp + 1, xv.y * w);
    atomicAdd(dp + 2, xv.z * w);
    atomicAdd(dp + 3, xv.w * w);
    if (count_deg && c == 0) atomicAdd(deg + d, 1.0f);
}

// agg[n][:] /= max(deg[n], 1)
__global__ void normalize_agg(float* __restrict__ agg,
                              const float* __restrict__ deg, int N)
{
    int tid = blockIdx.x * blockDim.x + threadIdx.x;
    if (tid >= N * DIM) return;
    agg[tid] = agg[tid] / fmaxf(deg[tid >> 7], 1.0f);
}

// ---------------------------------------------------------------------------
// Fused dual GEMM: out = act( A @ Wa + B @ Wb + bias ), A,B: [nrows,128],
// Wa,Wb: [128,128] row-major. Persistent waves: wave `wid` owns column tile
// ct=wid, preloads its weight fragments into registers ONCE (reused across all
// row tiles), and the block grid-strides over 16-row tiles. Each iteration
// stages the shared A/B row tile into LDS (coalesced b128), then the K-loop
// is pure ds_load_b64 + v_wmma_f32_16x16x4_f32.
// ---------------------------------------------------------------------------
__global__ void __launch_bounds__(256)
gemm_dual(const float* __restrict__ A, const float* __restrict__ Wa,
          const float* __restrict__ B, const float* __restrict__ Wb,
          const float* __restrict__ bias, float* __restrict__ out,
          int nrows, int relu)
{
    __shared__ float ldsA[16 * LDS_STRIDE];
    __shared__ float ldsB[16 * LDS_STRIDE];

    const int lane  = threadIdx.x & 31;
    const int wid   = threadIdx.x >> 5;     // 8 waves = 8 column tiles (DIM/16)
    const int mlo   = lane & 15;
    const int hi    = lane >> 4;
    const int klane = hi * 2;
    const int n     = wid * 16 + mlo;       // this wave's output column
    const int RT    = (nrows + 15) >> 4;

    // Register-resident weight fragments for this wave's column tile.
    v2f_t wa[32], wb[32];
#pragma unroll
    for (int k4 = 0; k4 < 32; ++k4) {
        const int k = k4 * 4 + klane;
        wa[k4].x = Wa[k * DIM + n];       wa[k4].y = Wa[(k + 1) * DIM + n];
        wb[k4].x = Wb[k * DIM + n];       wb[k4].y = Wb[(k + 1) * DIM + n];
    }
    const float bv = bias[n];

    for (int rt = blockIdx.x; rt < RT; rt += gridDim.x) {
        // Cooperative stage of the 16-row A and B tiles into LDS.
#pragma unroll
        for (int j = 0; j < 2; ++j) {
            const int c   = threadIdx.x + j * 256;   // 512 float4 chunks/matrix
            const int row = c >> 5;
            const int col = (c & 31) * 4;
            int gr = rt * 16 + row;
            if (gr > nrows - 1) gr = nrows - 1;      // clamp (tail tile)
            const float4 va = *(const float4*)(A + (size_t)gr * DIM + col);
            const float4 vb = *(const float4*)(B + (size_t)gr * DIM + col);
            *(float4*)&ldsA[row * LDS_STRIDE + col] = va;
            *(float4*)&ldsB[row * LDS_STRIDE + col] = vb;
        }
        __syncthreads();

        v8f_t acc;
#pragma unroll
        for (int i = 0; i < 8; ++i) acc[i] = bv;

#pragma unroll
        for (int k4 = 0; k4 < 32; ++k4) {
            const int off = mlo * LDS_STRIDE + k4 * 4 + klane;
            v2f_t a = *(const v2f_t*)&ldsA[off];
            acc = __builtin_amdgcn_wmma_f32_16x16x4_f32(false, a, false, wa[k4],
                                                        (short)0, acc, false, false);
            v2f_t b = *(const v2f_t*)&ldsB[off];
            acc = __builtin_amdgcn_wmma_f32_16x16x4_f32(false, b, false, wb[k4],
                                                        (short)0, acc, false, false);
        }

        // Epilogue: unguarded fast path for full tiles (the common case).
        if (rt * 16 + 16 <= nrows) {
            float* orow = out + ((size_t)rt * 16 + hi * 8) * DIM + n;
#pragma unroll
            for (int i = 0; i < 8; ++i) {
                float v = acc[i];
                if (relu) v = fmaxf(v, 0.0f);
                orow[(size_t)i * DIM] = v;
            }
        } else {
#pragma unroll
            for (int i = 0; i < 8; ++i) {
                const int r = rt * 16 + hi * 8 + i;
                if (r < nrows) {
                    float v = acc[i];
                    if (relu) v = fmaxf(v, 0.0f);
                    out[(size_t)r * DIM + n] = v;
                }
            }
        }
        __syncthreads();   // protect LDS before next iteration's fill
    }
}

// ---------------------------------------------------------------------------
// Fused decoder: per 16-edge tile, gather z[src], z[dst]; 16x64 hidden via
// WMMA over K=256 (two 128-chains); the "+1" ADP feature and bd1 folded into
// the accumulator init as a rank-1 term; ReLU; 64->1 projection by per-lane
// multiply + shfl_xor tree (width 16 matches the C-layout lane halves).
// ---------------------------------------------------------------------------
__global__ void decoder(const float* __restrict__ z, const int* __restrict__ eli,
                        const float* __restrict__ ew, const float* __restrict__ Wd1,
                        const float* __restrict__ bd1, const float* __restrict__ Wd2,
                        const float* __restrict__ bd2, float* __restrict__ out,
                        int nel)
{
    const int lane = threadIdx.x & 31;
    const int wid  = threadIdx.x >> 5;
    const int ntiles = (nel + 15) >> 4;
    const int tile = blockIdx.x * (blockDim.x >> 5) + wid;
    if (tile >= ntiles) return;

    const int mlo   = lane & 15;
    const int hi    = lane >> 4;
    const int klane = hi * 2;
    int row = tile * 16 + mlo;
    if (row > nel - 1) row = nel - 1;               // clamp, keep EXEC full
    const int isrc = eli[row];
    const int idst = eli[nel + row];

    v8f_t acc[4];
#pragma unroll
    for (int ct = 0; ct < 4; ++ct) {
        const int n = ct * 16 + mlo;
        const float b     = bd1[n];
        const float wlast = Wd1[256 * 64 + n];      // row for explicit_weight
#pragma unroll
        for (int i = 0; i < 8; ++i) {
            int r = tile * 16 + hi * 8 + i;
            if (r > nel - 1) r = nel - 1;
            acc[ct][i] = b + ew[r] * wlast;
        }
    }

    const float* zs = z + (size_t)isrc * DIM + klane;
    const float* zd = z + (size_t)idst * DIM + klane;
#pragma unroll 2
    for (int kk = 0; kk < DIM; kk += 4) {
        v2f_t a = *(const v2f_t*)(zs + kk);
#pragma unroll
        for (int ct = 0; ct < 4; ++ct) {
            const int n = ct * 16 + mlo;
            v2f_t w; w.x = Wd1[(kk + klane) * 64 + n];
                     w.y = Wd1[(kk + klane + 1) * 64 + n];
            acc[ct] = __builtin_amdgcn_wmma_f32_16x16x4_f32(false, a, false, w,
                                                            (short)0, acc[ct],
                                                            false, false);
        }
    }
#pragma unroll 2
    for (int kk = 0; kk < DIM; kk += 4) {
        v2f_t a = *(const v2f_t*)(zd + kk);
#pragma unroll
        for (int ct = 0; ct < 4; ++ct) {
            const int n = ct * 16 + mlo;
            v2f_t w; w.x = Wd1[(DIM + kk + klane) * 64 + n];
                     w.y = Wd1[(DIM + kk + klane + 1) * 64 + n];
            acc[ct] = __builtin_amdgcn_wmma_f32_16x16x4_f32(false, a, false, w,
                                                            (short)0, acc[ct],
                                                            false, false);
        }
    }

    float rowsum[8];
#pragma unroll
    for (int i = 0; i < 8; ++i) rowsum[i] = 0.0f;
#pragma unroll
    for (int ct = 0; ct < 4; ++ct) {
        const float w2 = Wd2[ct * 16 + mlo];
#pragma unroll
        for (int i = 0; i < 8; ++i) {
            float v = fmaxf(acc[ct][i], 0.0f) * w2;
            v += __shfl_xor(v, 1, 16);
            v += __shfl_xor(v, 2, 16);
            v += __shfl_xor(v, 4, 16);
            v += __shfl_xor(v, 8, 16);
            rowsum[i] += v;                          // full sum over n in group
        }
    }
    if (mlo == 0) {
        const float b2 = bd2[0];
        if (tile * 16 + 16 <= nel) {                 // full tile fast path
            float* op = out + tile * 16 + hi * 8;
#pragma unroll
            for (int i = 0; i < 8; ++i) op[i] = rowsum[i] + b2;
        } else {
#pragma unroll
            for (int i = 0; i < 8; ++i) {
                const int r = tile * 16 + hi * 8 + i;
                if (r < nel) out[r] = rowsum[i] + b2;
            }
        }
    }
}

// ---------------------------------------------------------------------------
extern "C" void kernel_launch(void* const* d_in, const int* in_sizes, int n_in,
                              void* d_out, int out_size, void* d_ws, size_t ws_size,
                              hipStream_t stream)
{
    (void)n_in; (void)out_size; (void)ws_size;
    const float* x        = (const float*)d_in[0];
    const int*   eidx     = (const int*)  d_in[1];
    const float* ewt      = (const float*)d_in[2];
    const int*   eli      = (const int*)  d_in[3];
    const float* expw     = (const float*)d_in[4];
    const float* W1_rel   = (const float*)d_in[5];
    const float* b1       = (const float*)d_in[6];
    const float* W1_root  = (const float*)d_in[7];
    const float* W2_rel   = (const float*)d_in[8];
    const float* b2       = (const float*)d_in[9];
    const float* W2_root  = (const float*)d_in[10];
    const float* Wd1      = (const float*)d_in[11];
    const float* bd1      = (const float*)d_in[12];
    const float* Wd2      = (const float*)d_in[13];
    const float* bd2      = (const float*)d_in[14];
    float* out = (float*)d_out;

    const int N  = in_sizes[0] / DIM;
    const int E  = in_sizes[2];
    const int EL = in_sizes[4];

    // Workspace: deg[N], agg[N*128] (reused across layers), h[N*128], z[N*128]
    float* deg = (float*)d_ws;
    float* agg = deg + N;
    float* h   = agg + (size_t)N * DIM;
    float* zz  = h   + (size_t)N * DIM;

    const int TB = 256;                              // 8 wave32 per block
    const long long sthreads = (long long)E * 32;
    const int sblocks = (int)((sthreads + TB - 1) / TB);
    const int nthreads = N * DIM;
    const int nblocks  = (nthreads + TB - 1) / TB;
    const int RT       = (N + 15) / 16;
    int gblocks        = RT < 1024 ? RT : 1024;      // persistent-wave grid
    const int dtiles   = (EL + 15) / 16;
    const int dblocks  = (dtiles + 7) / 8;

    // ---- layer 1: agg = scatter_mean(x[src]*w), h = relu(agg@W1_rel + b1 + x@W1_root)
    hipMemsetAsync(deg, 0, (size_t)N * sizeof(float), stream);
    hipMemsetAsync(agg, 0, (size_t)N * DIM * sizeof(float), stream);
    scatter_edges<<<sblocks, TB, 0, stream>>>(x, eidx, ewt, agg, deg, E, 1);
    normalize_agg<<<nblocks, TB, 0, stream>>>(agg, deg, N);
    gemm_dual<<<gblocks, TB, 0, stream>>>(agg, W1_rel, x, W1_root, b1, h, N, 1);

    // ---- layer 2: z = agg2@W2_rel + b2 + h@W2_root (agg buffer reused)
    hipMemsetAsync(agg, 0, (size_t)N * DIM * sizeof(float), stream);
    scatter_edges<<<sblocks, TB, 0, stream>>>(h, eidx, ewt, agg, deg, E, 0);
    normalize_agg<<<nblocks, TB, 0, stream>>>(agg, deg, N);
    gemm_dual<<<gblocks, TB, 0, stream>>>(agg, W2_rel, h, W2_root, b2, zz, N, 0);

    // ---- fused decoder -> out[EL]
    decoder<<<dblocks, TB, 0, stream>>>(zz, eli, expw, Wd1, bd1, Wd2, bd2, out, EL);
}